// LocalAttention_9423158247859
// MI455X (gfx1250) — compile-verified
//
#include <hip/hip_runtime.h>
#include <cstdint>
#include <cstddef>

// ---------------------------------------------------------------------------
// Local (banded) attention, B=2 L=4096 H=256 W=128, full f32 via
// V_WMMA_F32_16X16X4_F32.  Memory-bound on the 128MB dense attn output, so
// fp32 matrix path costs nothing vs fp8/bf16 and keeps reference precision.
// ---------------------------------------------------------------------------

typedef __attribute__((ext_vector_type(2))) float v2f;
typedef __attribute__((ext_vector_type(4))) float f4v;
typedef __attribute__((ext_vector_type(8))) float v8f;

#define BATCH   2
#define SEQ     4096
#define DIM     256
#define WBAND   128
#define NKT     17          // key tiles covering the band of a 16-row q block
#define SS_COLS (NKT * 16)  // 272
#define SS_PITCH 276        // 272 + 4 pad  (276 % 64 == 20 -> conflict-free)
#define QS_PITCH 260        // 256 + 4 pad

// D = A(16x4) * B(4x16) + C, all f32, wave32.
// Per-lane fragments (ISA 7.12.2):
//   A: lanes 0-15 hold row=lane, K={0,1}; lanes 16-31 row=lane-16, K={2,3}
//   B: lanes 0-15 hold col=lane, K={0,1}; lanes 16-31 col=lane-16, K={2,3}
//   C/D: VGPR r -> row r (lanes 0-15) / row r+8 (lanes 16-31), col = lane&15
__device__ __forceinline__ v8f wmma_f32_4(v2f a, v2f b, v8f c) {
    return __builtin_amdgcn_wmma_f32_16x16x4_f32(
        /*neg_a=*/false, a, /*neg_b=*/false, b,
        /*c_mod=*/(short)0, c, /*reuse_a=*/false, /*reuse_b=*/false);
}

// ---------------------------------------------------------------------------
// Y[m, n] = X[m, :256] @ Wm[:256, n] + bias[n]
// One wave per 16x16 tile; 4 waves per block cover 64 output columns.
// ---------------------------------------------------------------------------
__global__ __launch_bounds__(128)
void la_proj_gemm(const float* __restrict__ X, const float* __restrict__ Wm,
                  const float* __restrict__ bias, float* __restrict__ Y) {
    const int lane  = threadIdx.x & 31;
    const int wave  = threadIdx.x >> 5;
    const int mt    = blockIdx.y;
    const int nt    = blockIdx.x * 4 + wave;
    const int l15   = lane & 15;
    const int khalf = (lane >> 4) << 1;   // 0 or 2
    const int row   = mt * 16 + l15;      // A row for this lane
    const int col   = nt * 16 + l15;      // B/C column for this lane

    const float* xrow = X + (size_t)row * DIM;
    v8f acc = {};
    #pragma unroll 8
    for (int k = 0; k < DIM; k += 4) {
        const int kk = k + khalf;
        v2f a = *(const v2f*)(xrow + kk);                       // contiguous 8B
        v2f b;
        b.x = Wm[(size_t)kk * DIM + col];
        b.y = Wm[(size_t)(kk + 1) * DIM + col];
        acc = wmma_f32_4(a, b, acc);
    }
    const float bv    = bias[col];
    const int   mbase = mt * 16 + ((lane >> 4) << 3);
    #pragma unroll
    for (int r = 0; r < 8; ++r)
        Y[(size_t)(mbase + r) * DIM + col] = acc[r] + bv;
}

// ---------------------------------------------------------------------------
// Non-temporal zero of the dense attn tensor (written once, never re-read).
// ---------------------------------------------------------------------------
__global__ void la_zero_attn(f4v* __restrict__ p, size_t n4) {
    size_t i      = (size_t)blockIdx.x * blockDim.x + threadIdx.x;
    size_t stride = (size_t)gridDim.x * blockDim.x;
    f4v z = {0.f, 0.f, 0.f, 0.f};
    for (; i < n4; i += stride)
        __builtin_nontemporal_store(z, p + i);
}

// ---------------------------------------------------------------------------
// One 128-thread block (4 waves) per (batch, 16-row q block):
//   scores (WMMA, band-masked) -> softmax -> attn write -> context (WMMA)
// ---------------------------------------------------------------------------
__global__ __launch_bounds__(128)
void la_local_attn(const float* __restrict__ Q, const float* __restrict__ K,
                   const float* __restrict__ V, float* __restrict__ ctx,
                   float* __restrict__ attn) {
    __shared__ float qs[16][QS_PITCH];
    __shared__ float ss[16][SS_PITCH];

    const int tid  = threadIdx.x;
    const int lane = tid & 31;
    const int wave = tid >> 5;
    const int b    = blockIdx.x >> 8;     // / (SEQ/16)
    const int qb   = blockIdx.x & 255;
    const size_t rowbase = (size_t)b * SEQ + (size_t)qb * 16;

    const int l15   = lane & 15;
    const int khalf = (lane >> 4) << 1;

    // ---- stage q tile [16 x 256] into LDS -------------------------------
    for (int idx = tid; idx < 16 * DIM; idx += 128) {
        const int r = idx >> 8, c = idx & 255;
        qs[r][c] = Q[(rowbase + r) * DIM + c];
    }
    // ---- init score window to -inf (covers out-of-range / masked) -------
    for (int idx = tid; idx < 16 * SS_COLS; idx += 128)
        ss[idx / SS_COLS][idx % SS_COLS] = -INFINITY;
    __syncthreads();

    // ---- scores: s = q . k^T over the 17-tile band (wave-uniform guards) -
    for (int t = wave; t < NKT; t += 4) {
        const int jb = qb - 8 + t;
        if (jb < 0 || jb >= SEQ / 16) continue;          // uniform per wave
        const float* krow = K + ((size_t)b * SEQ + (size_t)jb * 16 + l15) * DIM;
        v8f acc = {};
        #pragma unroll 8
        for (int k = 0; k < DIM; k += 4) {
            const int kk = k + khalf;
            v2f a = *(const v2f*)(&qs[l15][kk]);          // 8B aligned (pitch%2==0)
            v2f bb = *(const v2f*)(krow + kk);            // B[kk][col]=K[j0+col][kk]
            acc = wmma_f32_4(a, bb, acc);
        }
        const int j = jb * 16 + l15;                      // this lane's key index
        #pragma unroll
        for (int r = 0; r < 8; ++r) {
            const int m = r + ((lane >> 4) << 3);
            const int d = (qb * 16 + m) - j;
            if (d <= WBAND && d >= -WBAND)
                ss[m][t * 16 + l15] = acc[r];             // else stays -inf
        }
    }
    __syncthreads();

    // ---- softmax: 8 lanes per row, shuffle reductions -------------------
    {
        const int r = tid >> 3, sub = tid & 7;
        float mx = -INFINITY;
        for (int c = sub; c < SS_COLS; c += 8) mx = fmaxf(mx, ss[r][c]);
        for (int off = 1; off < 8; off <<= 1) mx = fmaxf(mx, __shfl_xor(mx, off, 32));
        float sum = 0.f;
        for (int c = sub; c < SS_COLS; c += 8) {
            const float e = __expf(ss[r][c] - mx);        // exp(-inf)=0
            ss[r][c] = e;
            sum += e;
        }
        for (int off = 1; off < 8; off <<= 1) sum += __shfl_xor(sum, off, 32);
        const float inv = 1.0f / sum;                     // diag guarantees sum>0
        const size_t arow = (rowbase + r) * (size_t)SEQ;
        for (int c = sub; c < SS_COLS; c += 8) {
            const float p = ss[r][c] * inv;
            ss[r][c] = p;
            const int j = (qb - 8) * 16 + c;
            if (j >= 0 && j < SEQ)
                __builtin_nontemporal_store(p, &attn[arow + j]);
        }
    }
    __syncthreads();

    // ---- context = attn[16 x 272] @ v[272 x 256]; wave owns 64 columns --
    const int h0 = wave * 64;
    v8f acc[4] = {};
    for (int t = 0; t < NKT; ++t) {
        const int jb = qb - 8 + t;
        if (jb < 0 || jb >= SEQ / 16) continue;           // attn there is 0
        const float* vbase = V + ((size_t)b * SEQ + (size_t)jb * 16) * DIM;
        #pragma unroll
        for (int ks = 0; ks < 4; ++ks) {
            const int kl = ks * 4 + khalf;                // local K in tile
            v2f a = *(const v2f*)(&ss[l15][t * 16 + kl]); // A[m][k]=attn
            #pragma unroll
            for (int nt = 0; nt < 4; ++nt) {
                const int col = h0 + nt * 16 + l15;
                v2f bb;
                bb.x = vbase[(size_t)kl * DIM + col];
                bb.y = vbase[(size_t)(kl + 1) * DIM + col];
                acc[nt] = wmma_f32_4(a, bb, acc[nt]);
            }
        }
    }
    #pragma unroll
    for (int nt = 0; nt < 4; ++nt) {
        const int col = h0 + nt * 16 + l15;
        #pragma unroll
        for (int r = 0; r < 8; ++r) {
            const int m = r + ((lane >> 4) << 3);
            ctx[(rowbase + m) * DIM + col] = acc[nt][r];
        }
    }
}

// ---------------------------------------------------------------------------
extern "C" void kernel_launch(void* const* d_in, const int* in_sizes, int n_in,
                              void* d_out, int out_size, void* d_ws, size_t ws_size,
                              hipStream_t stream) {
    (void)in_sizes; (void)n_in; (void)out_size; (void)ws_size;
    const float* query = (const float*)d_in[0];
    const float* key   = (const float*)d_in[1];
    const float* value = (const float*)d_in[2];
    const float* Wq    = (const float*)d_in[3];
    const float* bq    = (const float*)d_in[4];
    const float* Wk    = (const float*)d_in[5];
    const float* bk    = (const float*)d_in[6];
    const float* Wv    = (const float*)d_in[7];
    const float* bv    = (const float*)d_in[8];
    const float* Wo    = (const float*)d_in[9];
    const float* bo    = (const float*)d_in[10];

    const size_t nTok = (size_t)BATCH * SEQ;             // 8192 rows
    float* out  = (float*)d_out;                          // [B,L,256]
    float* attn = out + nTok * DIM;                       // [B,L,L]

    float* qp = (float*)d_ws;                             // 8 MB each
    float* kp = qp + nTok * DIM;
    float* vp = kp + nTok * DIM;
    float* cx = vp + nTok * DIM;

    const dim3 gg(DIM / 64, (unsigned)(nTok / 16));       // (4, 512) x 128 thr
    la_proj_gemm<<<gg, 128, 0, stream>>>(query, Wq, bq, qp);
    la_proj_gemm<<<gg, 128, 0, stream>>>(key,   Wk, bk, kp);
    la_proj_gemm<<<gg, 128, 0, stream>>>(value, Wv, bv, vp);

    const size_t n4 = (size_t)BATCH * SEQ * SEQ / 4;      // attn float4 count
    la_zero_attn<<<4096, 256, 0, stream>>>((f4v*)attn, n4);

    la_local_attn<<<BATCH * (SEQ / 16), 128, 0, stream>>>(qp, kp, vp, cx, attn);

    la_proj_gemm<<<gg, 128, 0, stream>>>(cx, Wo, bo, out);
}